// EnergyGCN_49340584296527
// MI455X (gfx1250) — compile-verified
//
#include <hip/hip_runtime.h>

// ---------------------------------------------------------------------------
// EnergyGCN forward for MI455X (gfx1250, wave32, WMMA).
//
// Sizes (fixed by the reference):
#define N_NODES  50000
#define N_EDGES  800000
#define DIM_IN   256
#define DIM_H    128
#define DIM_OUT  64
#define N_LAYERS 4
// ---------------------------------------------------------------------------

typedef float v2f __attribute__((ext_vector_type(2)));
typedef float v8f __attribute__((ext_vector_type(8)));

// D = A(16x4 f32) * B(4x16 f32) + C(16x16 f32), one matrix per wave32.
__device__ __forceinline__ v8f wmma_f32_16x16x4(v2f a, v2f b, v8f c) {
  // 8 args: (neg_a, A, neg_b, B, c_mod, C, reuse_a, reuse_b)
  return __builtin_amdgcn_wmma_f32_16x16x4_f32(false, a, false, b, (short)0, c,
                                               false, false);
}

// ---------------------------------------------------------------------------
// Tiled fp32 WMMA GEMM.  One block = 256 threads = 8 waves, computes a
// BM(=64) x BN tile where BN is the *entire* output width (128 or 64), so the
// grid only tiles M.  K is staged through LDS in chunks of 16.
//
// LDS layout holds K-PAIRS: lAp[kp][m] = {A(2kp), A(2kp+1)} so that each WMMA
// fragment (lane needs elements k,k+1) is a single contiguous ds_load_b64
// into an even-aligned VGPR pair — no v_mov patch-up before v_wmma.
//
//   wave w: row-tile tm = w&3 (A fragment shared across its NTILE col tiles),
//           col-tiles tn0..tn0+NTILE-1 with tn0 = (w>>2)*NTILE.
// TRANS_B: B is [BN x K] row-major and used transposed (h @ Wr.T).
// ---------------------------------------------------------------------------
template <int BN, int NTILE, bool TRANS_B, bool RELU, bool BIAS, bool COPY>
__global__ __launch_bounds__(256) void wmma_gemm(
    const float* __restrict__ A, const float* __restrict__ B,
    const float* __restrict__ bias, float* __restrict__ C,
    float* __restrict__ C2, int M, int K) {
  constexpr int BM = 64;
  __shared__ v2f lAp[8][BM];   // k-pair kp in [0,8), m in [0,BM)
  __shared__ v2f lBp[8][BN];   // k-pair kp in [0,8), n in [0,BN)

  const int tid  = threadIdx.x;
  const int lane = tid & 31;
  const int wid  = tid >> 5;
  const int m0   = blockIdx.x * BM;

  const int tm  = wid & 3;                // 4 row tiles of 16
  const int tn0 = (wid >> 2) * NTILE;     // column tile base
  const int hb  = lane >> 4;              // half-wave: 0 -> K even pair, 1 -> +2

  v8f c[NTILE];
#pragma unroll
  for (int i = 0; i < NTILE; ++i) c[i] = {};

  const int nChunks = K >> 4;
  for (int ch = 0; ch < nChunks; ++ch) {
    const int k0 = ch << 4;

    // ---- stage A tile: float4 along K -> two k-pair stores ----
    {
      const int m  = tid >> 2;
      const int kq = (tid & 3) << 2;      // 0,4,8,12
      int gm = m0 + m;
      if (gm >= M) gm = M - 1;  // tail: compute garbage, store is guarded
      const float4 v = *(const float4*)&A[(size_t)gm * K + k0 + kq];
      lAp[(kq >> 1) + 0][m] = v2f{v.x, v.y};
      lAp[(kq >> 1) + 1][m] = v2f{v.z, v.w};
    }
    // ---- stage B tile ----
    if constexpr (!TRANS_B) {
      // lBp[kp][n] = {B[(k0+2kp)*BN+n], B[(k0+2kp+1)*BN+n]}
      constexpr int CH = 8 * (BN / 4);    // each chunk: 4 n's at one kp
#pragma unroll
      for (int c4 = tid; c4 < CH; c4 += 256) {
        const int kp = c4 / (BN / 4);
        const int n4 = (c4 % (BN / 4)) << 2;
        const float4 r0 =
            *(const float4*)&B[(size_t)(k0 + 2 * kp + 0) * BN + n4];
        const float4 r1 =
            *(const float4*)&B[(size_t)(k0 + 2 * kp + 1) * BN + n4];
        lBp[kp][n4 + 0] = v2f{r0.x, r1.x};
        lBp[kp][n4 + 1] = v2f{r0.y, r1.y};
        lBp[kp][n4 + 2] = v2f{r0.z, r1.z};
        lBp[kp][n4 + 3] = v2f{r0.w, r1.w};
      }
    } else {
      // lBp[kp][n] = {B[n*K + k0+2kp], B[n*K + k0+2kp+1]} : pairs come free
      constexpr int CH = (16 * BN) / 4;
#pragma unroll
      for (int c4 = tid; c4 < CH; c4 += 256) {
        const int n  = c4 >> 2;
        const int kq = (c4 & 3) << 2;
        const float4 v = *(const float4*)&B[(size_t)n * K + k0 + kq];
        lBp[(kq >> 1) + 0][n] = v2f{v.x, v.y};
        lBp[(kq >> 1) + 1][n] = v2f{v.z, v.w};
      }
    }
    __syncthreads();

    // ---- 4 WMMA k-steps per 16-wide chunk ----
#pragma unroll
    for (int ks = 0; ks < 4; ++ks) {
      // ISA layout: lanes 0-15 hold K = 4ks+{0,1}, lanes 16-31 K = 4ks+{2,3}
      const int kp = (ks << 1) + hb;
      const v2f a = lAp[kp][(tm << 4) + (lane & 15)];
#pragma unroll
      for (int i = 0; i < NTILE; ++i) {
        const v2f b = lBp[kp][((tn0 + i) << 4) + (lane & 15)];
        c[i] = wmma_f32_16x16x4(a, b, c[i]);
      }
    }
    __syncthreads();
  }

  // ---- epilogue: C VGPR r holds M=r (lanes 0-15) / M=8+r (lanes 16-31) ----
  const int mhalf = hb << 3;
#pragma unroll
  for (int i = 0; i < NTILE; ++i) {
    const int gn = ((tn0 + i) << 4) + (lane & 15);
    float bv = 0.0f;
    if constexpr (BIAS) bv = bias[gn];
#pragma unroll
    for (int r = 0; r < 8; ++r) {
      const int gm = m0 + (tm << 4) + mhalf + r;
      if (gm < M) {
        float v = c[i][r] + bv;
        if constexpr (RELU) v = fmaxf(v, 0.0f);
        C[(size_t)gm * BN + gn] = v;
        if constexpr (COPY) C2[(size_t)gm * BN + gn] = v;
      }
    }
  }
}

// ---------------------------------------------------------------------------
// Small helpers
// ---------------------------------------------------------------------------
__global__ void fill_zero(float* __restrict__ p, long long n) {
  const long long i = (long long)blockIdx.x * blockDim.x + threadIdx.x;
  if (i < n) p[i] = 0.0f;
}

// deg[row[e]] += 1 over real edges; the self-loop "+1" is folded into
// make_dinv, so the column side never needs counting (matches reference:
// deg is segment_sum over `row` only).
__global__ void deg_accum(const long long* __restrict__ row,
                          float* __restrict__ deg, int E) {
  const int e = blockIdx.x * blockDim.x + threadIdx.x;
  if (e < E) atomicAdd(&deg[(int)row[e]], 1.0f);
}

__global__ void make_dinv(float* __restrict__ d, int n) {
  const int i = blockIdx.x * blockDim.x + threadIdx.x;
  if (i < n) d[i] = rsqrtf(d[i] + 1.0f);  // +1 self loop => always > 0
}

// One wave per edge; lane handles 4 of the 128 channels.  t (25.6 MB) and agg
// are L2-resident, so gathers + float atomics run at L2 speed, not HBM.
__global__ __launch_bounds__(256) void edge_scatter(
    const long long* __restrict__ ei, const float* __restrict__ dinv,
    const float* __restrict__ t, float* __restrict__ agg, int E) {
  const int g    = blockIdx.x * blockDim.x + threadIdx.x;
  const int e    = g >> 5;
  const int lane = g & 31;
  if (e >= E) return;
  if (lane == 0 && e + 4096 < E) {  // stream the edge list ahead of us
    __builtin_prefetch(&ei[e + 4096], 0, 0);
    __builtin_prefetch(&ei[(size_t)E + e + 4096], 0, 0);
  }
  const int   r  = (int)ei[e];               // destination (row)
  const int   cc = (int)ei[(size_t)E + e];   // source (col)
  const float w  = dinv[r] * dinv[cc];
  const float4 v = *(const float4*)&t[(size_t)cc * DIM_H + (lane << 2)];
  float* dst = &agg[(size_t)r * DIM_H + (lane << 2)];
  atomicAdd(dst + 0, w * v.x);
  atomicAdd(dst + 1, w * v.y);
  atomicAdd(dst + 2, w * v.z);
  atomicAdd(dst + 3, w * v.w);
}

// h = 6t - 3(agg + dinv^2 * t);  h0 = (1+tanh(eps))h0 + h;  h = relu(h0)
__global__ void combine(const float* __restrict__ t,
                        const float* __restrict__ agg,
                        const float* __restrict__ dinv,
                        const float* __restrict__ eps_l,
                        float* __restrict__ h0, float* __restrict__ h,
                        long long total) {
  const long long idx = (long long)blockIdx.x * blockDim.x + threadIdx.x;
  if (idx >= total) return;
  const int   node = (int)(idx >> 7);  // / DIM_H
  const float g    = 1.0f + tanhf(eps_l[0]);
  const float di   = dinv[node];
  const float tv   = t[idx];
  const float a    = agg[idx] + di * di * tv;  // add self-loop term
  const float v    = 6.0f * tv - 3.0f * a;
  const float nh   = g * h0[idx] + v;
  h0[idx] = nh;
  h[idx]  = fmaxf(nh, 0.0f);  // safe: reads agg[idx] (==h[idx]) above
}

// ---------------------------------------------------------------------------
extern "C" void kernel_launch(void* const* d_in, const int* in_sizes, int n_in,
                              void* d_out, int out_size, void* d_ws,
                              size_t ws_size, hipStream_t stream) {
  (void)in_sizes; (void)n_in; (void)out_size; (void)ws_size;
  const float*     x   = (const float*)d_in[0];
  const float*     W1  = (const float*)d_in[1];
  const float*     b1  = (const float*)d_in[2];
  const float*     Wr  = (const float*)d_in[3];
  const float*     eps = (const float*)d_in[4];
  const float*     W2  = (const float*)d_in[5];
  const float*     b2  = (const float*)d_in[6];
  const long long* ei  = (const long long*)d_in[7];  // int64 edge_index [2,E]
  float*           out = (float*)d_out;

  // Workspace: dinv[N] | h[N*H] | h0[N*H] | t[N*H]; agg aliases h (h is dead
  // between the right-GEMM that consumed it and the combine that rewrites it).
  float* ws   = (float*)d_ws;
  float* dinv = ws;
  float* h    = ws + N_NODES;
  float* h0   = h + (size_t)N_NODES * DIM_H;
  float* t    = h0 + (size_t)N_NODES * DIM_H;
  float* agg  = h;

  const long long NH = (long long)N_NODES * DIM_H;

  // --- degrees -> dinv (self loops folded in analytically) ---
  fill_zero<<<(N_NODES + 255) / 256, 256, 0, stream>>>(dinv, N_NODES);
  deg_accum<<<(N_EDGES + 255) / 256, 256, 0, stream>>>(ei, dinv, N_EDGES);
  make_dinv<<<(N_NODES + 255) / 256, 256, 0, stream>>>(dinv, N_NODES);

  // --- h = relu(x @ W1 + b1); h0 = h (fused copy) ---
  wmma_gemm<DIM_H, 4, false, true, true, true>
      <<<(N_NODES + 63) / 64, 256, 0, stream>>>(x, W1, b1, h, h0, N_NODES,
                                                DIM_IN);

  for (int l = 0; l < N_LAYERS; ++l) {
    // t = h @ Wr[l].T
    wmma_gemm<DIM_H, 4, true, false, false, false>
        <<<(N_NODES + 63) / 64, 256, 0, stream>>>(
            h, Wr + (size_t)l * DIM_H * DIM_H, nullptr, t, nullptr, N_NODES,
            DIM_H);
    // agg = Ahat_edges @ t  (edge part; self-loop term added in combine)
    fill_zero<<<(int)((NH + 255) / 256), 256, 0, stream>>>(agg, NH);
    edge_scatter<<<(N_EDGES * 32 + 255) / 256, 256, 0, stream>>>(ei, dinv, t,
                                                                 agg, N_EDGES);
    // h0 = (1+tanh(eps))h0 + (6t - 3*Ahat t);  h = relu(h0)
    combine<<<(int)((NH + 255) / 256), 256, 0, stream>>>(t, agg, dinv, eps + l,
                                                         h0, h, NH);
  }

  // --- out = h @ W2 + b2 ---
  wmma_gemm<DIM_OUT, 2, false, false, true, false>
      <<<(N_NODES + 63) / 64, 256, 0, stream>>>(h, W2, b2, out, nullptr,
                                                N_NODES, DIM_H);
}